// Decoder_20547123544511
// MI455X (gfx1250) — compile-verified
//
#include <hip/hip_runtime.h>
#include <hip/hip_bf16.h>

// ---------------------------------------------------------------------------
// Tacotron2-style decoder for MI455X (gfx1250).
// fp32 WMMA (16x16x4) for all GEMM-shaped work; async global->LDS staging
// (ASYNCcnt) for shared A-operands; prefetch for L2-resident weight streams.
// ---------------------------------------------------------------------------

#define BB 32
#define TT 256
#define LL 256
#define IDIM 512
#define ODIM 80
#define DUNITS 512
#define PUNITS 256
#define ATT_DIM 128
#define ACONV_CHANS 32
#define ACONV_K 31
#define PN_CHANS 512
#define PN_K 5
#define ZONEOUT 0.1f
#define BN_INV 0.99999500003749969f  // 1/sqrt(1+1e-5)
#define KC 128                       // K-chunk for LDS double buffering

typedef __attribute__((ext_vector_type(2))) float v2f;
typedef __attribute__((ext_vector_type(8))) float v8f;
typedef __attribute__((ext_vector_type(4))) int v4i;

#ifndef __has_builtin
#define __has_builtin(x) 0
#endif

#if __has_builtin(__builtin_amdgcn_global_load_async_to_lds_b128)
#define HAVE_ASYNC_LDS 1
#else
#define HAVE_ASYNC_LDS 0
#endif

#if __has_builtin(__builtin_amdgcn_s_wait_asynccnt)
#define WAIT_ASYNC(n) __builtin_amdgcn_s_wait_asynccnt(n)
#else
#define WAIT_ASYNC(n) asm volatile("s_wait_asynccnt %0" ::"i"(n) : "memory")
#endif

// 16-byte global -> LDS copy; async (ASYNCcnt-tracked) when available.
__device__ __forceinline__ void async_cp16(const float* g, float* l) {
#if HAVE_ASYNC_LDS
    __builtin_amdgcn_global_load_async_to_lds_b128((v4i*)(void*)g, (v4i*)l, 0, 0);
#else
    *(float4*)l = *(const float4*)g;
#endif
}

// workspace layout (float offsets)
#define WS_ENCPROJ 0            // B*T*128        = 1048576
#define WS_Z0 1048576           // 2 * B*512      = 32768
#define WS_C0 1081344
#define WS_Z1 1114112
#define WS_C1 1146880
#define WS_AW 1179648           // 2 * B*T        = 16384
#define WS_X  1196032           // B*768          = 24576
#define WS_H0 1220608           // B*512*256      = 4194304
#define WS_H1 5414912           // B*512*256      = 4194304

// output layout (float offsets)
#define OUT_AFTER 0             // B*L*ODIM = 655360
#define OUT_BEFORE 655360
#define OUT_LOGITS 1310720      // B*L = 8192
#define OUT_AWS 1318912         // B*L*T = 2097152

__device__ __forceinline__ v8f wmma_f32(v2f a, v2f b, v8f c) {
    return __builtin_amdgcn_wmma_f32_16x16x4_f32(false, a, false, b, (short)0, c,
                                                 false, false);
}

__device__ __forceinline__ float sigmoidf(float x) {
    return 1.0f / (1.0f + __expf(-x));
}

// ---------------------------------------------------------------------------
// init: zero state slot 0, uniform attention weights
// ---------------------------------------------------------------------------
__global__ void k_init(float* ws, const int* ilens) {
    int i = blockIdx.x * blockDim.x + threadIdx.x;  // 0 .. B*512-1 (16384)
    ws[WS_Z0 + i] = 0.0f;
    ws[WS_C0 + i] = 0.0f;
    ws[WS_Z1 + i] = 0.0f;
    ws[WS_C1 + i] = 0.0f;
    if (i < BB * TT) {
        int b = i >> 8, t = i & 255;
        int il = ilens[b];
        ws[WS_AW + i] = (t < il) ? (1.0f / (float)il) : 0.0f;
    }
}

// ---------------------------------------------------------------------------
// enc_proj = enc_z @ W_enc + b_enc   (8192 x 128, K=512), WMMA f32 16x16x4.
// Block = 8 waves sharing one 16-row A tile, staged via async->LDS.
// grid: 512 blocks * 256 threads.
// ---------------------------------------------------------------------------
__global__ void k_enc_proj(const float* __restrict__ enc_z,
                           const float* __restrict__ W_enc,
                           const float* __restrict__ b_enc,
                           float* __restrict__ enc_proj) {
    __shared__ float sa[16][IDIM];  // 32 KB
    int tid = threadIdx.x;
    int lane = tid & 31, w = tid >> 5;
    int m0 = blockIdx.x * 16;
    int n0 = w * 16;  // 8 n-tiles across ATT_DIM
    int l16 = lane & 15, hi = lane >> 4;

    // stage A tile (16 x 512): 2048 x b128, 8 per thread
    for (int j = 0; j < 8; j++) {
        int f = tid + j * 256;
        int r = f >> 7;            // row 0..15
        int col = (f & 127) * 4;   // 0..508
        async_cp16(enc_z + (size_t)(m0 + r) * IDIM + col, &sa[r][col]);
    }
    WAIT_ASYNC(0);
    __syncthreads();

    v8f acc = {};
    for (int k = 0; k < IDIM; k += 4) {
        int kq = k + 2 * hi;
        v2f a; a.x = sa[l16][kq]; a.y = sa[l16][kq + 1];
        v2f bm; bm.x = W_enc[kq * ATT_DIM + n0 + l16];
        bm.y = W_enc[(kq + 1) * ATT_DIM + n0 + l16];
        acc = wmma_f32(a, bm, acc);
    }
    float bias = b_enc[n0 + l16];
    for (int v = 0; v < 8; v++) {
        int m = m0 + v + 8 * hi;
        enc_proj[m * ATT_DIM + n0 + l16] = acc[v] + bias;
    }
}

// ---------------------------------------------------------------------------
// per-step: location attention + softmax + context + prenet.
// grid: B blocks, 256 threads (one per encoder frame).
// ---------------------------------------------------------------------------
__global__ void k_att(const float* __restrict__ enc_z, const float* __restrict__ ys,
                      const int* __restrict__ ilens, const float* __restrict__ W_dec,
                      const float* __restrict__ loc_kernel,
                      const float* __restrict__ W_att, const float* __restrict__ g_w,
                      const float* __restrict__ g_b, const float* __restrict__ Wp0,
                      const float* __restrict__ bp0, const float* __restrict__ Wp1,
                      const float* __restrict__ bp1, float* ws,
                      float* __restrict__ out_aws, int t) {
    int b = blockIdx.x, tid = threadIdx.x;
    int s = t & 1;
    __shared__ float aw_prev[TT];
    __shared__ float dec[ATT_DIM];
    __shared__ float sh[TT];
    __shared__ float awn[TT];
    __shared__ float p0[PUNITS];

    const float* z0 = ws + WS_Z0 + s * (BB * DUNITS) + b * DUNITS;
    aw_prev[tid] = ws[WS_AW + s * (BB * TT) + b * TT + tid];
    if (tid < ATT_DIM) {
        float acc = 0.0f;
        for (int j = 0; j < DUNITS; j++) acc += z0[j] * W_dec[j * ATT_DIM + tid];
        dec[tid] = acc;
    }
    __syncthreads();

    // 1D location conv (31 taps, 32 channels), then energy for frame `tid`
    float loc[ACONV_CHANS];
    for (int c = 0; c < ACONV_CHANS; c++) {
        float a = 0.0f;
        for (int f = 0; f < ACONV_K; f++) {
            int p = tid + f - 15;
            if (p >= 0 && p < TT) a += aw_prev[p] * loc_kernel[c * ACONV_K + f];
        }
        loc[c] = a;
    }
    const float* ep = &ws[WS_ENCPROJ] + (size_t)(b * TT + tid) * ATT_DIM;
    float e = g_b[0];
    for (int d = 0; d < ATT_DIM; d++) {
        float v = dec[d] + ep[d];
        for (int c = 0; c < ACONV_CHANS; c++) v += loc[c] * W_att[c * ATT_DIM + d];
        e += tanhf(v) * g_w[d];
    }
    int il = ilens[b];
    float es = (tid < il) ? (2.0f * e) : -2.0e30f;  // SCALING * masked energy

    // masked softmax over T
    sh[tid] = es; __syncthreads();
    for (int o = 128; o > 0; o >>= 1) {
        if (tid < o) sh[tid] = fmaxf(sh[tid], sh[tid + o]);
        __syncthreads();
    }
    float mx = sh[0]; __syncthreads();
    float ex = __expf(es - mx);
    sh[tid] = ex; __syncthreads();
    for (int o = 128; o > 0; o >>= 1) {
        if (tid < o) sh[tid] += sh[tid + o];
        __syncthreads();
    }
    float a_n = ex / sh[0];
    awn[tid] = a_n;
    ws[WS_AW + (s ^ 1) * (BB * TT) + b * TT + tid] = a_n;
    out_aws[(size_t)(b * LL + t) * TT + tid] = a_n;
    __syncthreads();

    // context vector -> x[0:512]
    float* x = ws + WS_X + b * (IDIM + PUNITS);
    const float* ez = enc_z + (size_t)b * TT * IDIM;
    for (int d = tid; d < IDIM; d += 256) {
        float acc = 0.0f;
        for (int k = 0; k < TT; k++) acc += awn[k] * ez[k * IDIM + d];
        x[d] = acc;
    }

    // prenet (y_{t-1} -> 256 -> 256) -> x[512:768]
    const float* yprev = ys + (size_t)(b * LL + (t - 1)) * ODIM;
    float acc0 = bp0[tid];
    if (t > 0)
        for (int q = 0; q < ODIM; q++) acc0 += yprev[q] * Wp0[q * PUNITS + tid];
    p0[tid] = fmaxf(acc0, 0.0f);
    __syncthreads();
    float acc1 = bp1[tid];
    for (int q = 0; q < PUNITS; q++) acc1 += p0[q] * Wp1[q * PUNITS + tid];
    x[IDIM + tid] = fmaxf(acc1, 0.0f);
}

// ---------------------------------------------------------------------------
// LSTMCell: gates = [x1 | hprev] @ [Wih ; Whh] + bias  (32 x 2048), WMMA,
// A staged to LDS in double-buffered async K-chunks; fused cell + zoneout.
// grid: 32 blocks (16 hidden cols each), 256 threads = 8 waves
//   wave w: gate group g=w>>1 (i,f,g,o), m-tile mt=w&1.
// ---------------------------------------------------------------------------
__global__ void k_lstm(const float* __restrict__ x1, int K1,
                       const float* __restrict__ Wih,
                       const float* __restrict__ hprev,
                       const float* __restrict__ Whh,
                       const float* __restrict__ bih, const float* __restrict__ bhh,
                       const float* __restrict__ cprev, float* __restrict__ hout,
                       float* __restrict__ cout) {
    __shared__ float sa[2][BB][KC];   // 32 KB double-buffered A chunks
    __shared__ float gsh[4][BB][16];  // 8 KB gate staging
    int tid = threadIdx.x;
    int lane = tid & 31, w = tid >> 5;
    int g = w >> 1, mt = w & 1;
    int n0 = g * DUNITS + blockIdx.x * 16;  // column in 2048-wide gate matrix
    int l16 = lane & 15, hi = lane >> 4;
    int m0 = mt * 16;
    int Ktot = K1 + DUNITS;
    int nch = Ktot / KC;

    // stage chunk 0 (always inside x1: K1 >= KC)
    for (int j = 0; j < 4; j++) {
        int f = tid + j * 256;
        int r = f >> 5, col = (f & 31) * 4;
        async_cp16(x1 + (size_t)r * K1 + col, &sa[0][r][col]);
    }

    v8f acc = {};
    for (int c = 0; c < nch; c++) {
        int k0 = c * KC;
        if (c + 1 < nch) {
            int k0n = k0 + KC;
            const float* base;
            int ld;
            if (k0n < K1) { base = x1 + k0n; ld = K1; }
            else          { base = hprev + (k0n - K1); ld = DUNITS; }
            for (int j = 0; j < 4; j++) {
                int f = tid + j * 256;
                int r = f >> 5, col = (f & 31) * 4;
                async_cp16(base + (size_t)r * ld + col, &sa[(c + 1) & 1][r][col]);
            }
            // prefetch next chunk's weight rows at our column (L2-resident)
            const float* Wn = (k0n < K1) ? (Wih + (size_t)k0n * 2048)
                                         : (Whh + (size_t)(k0n - K1) * 2048);
            __builtin_prefetch(Wn + (size_t)(lane * 4) * 2048 + n0, 0, 0);
            WAIT_ASYNC(4);  // newest 4 may be in flight; current chunk complete
        } else {
            WAIT_ASYNC(0);
        }
        __syncthreads();

        const float* W = (k0 < K1) ? (Wih + (size_t)k0 * 2048)
                                   : (Whh + (size_t)(k0 - K1) * 2048);
        const float* sArow = &sa[c & 1][m0 + l16][0];
        for (int kk = 0; kk < KC; kk += 4) {
            int kq = kk + 2 * hi;
            v2f a; a.x = sArow[kq]; a.y = sArow[kq + 1];
            v2f bm; bm.x = W[(size_t)kq * 2048 + n0 + l16];
            bm.y = W[(size_t)(kq + 1) * 2048 + n0 + l16];
            acc = wmma_f32(a, bm, acc);
        }
        __syncthreads();  // buffer c&1 may be re-staged next iteration
    }

    float bias = bih[n0 + l16] + bhh[n0 + l16];
    for (int v = 0; v < 8; v++) gsh[g][m0 + v + 8 * hi][l16] = acc[v] + bias;
    __syncthreads();

    // cell update + zoneout for the 16 hidden units this block owns
    int j0 = blockIdx.x * 16;
    for (int i = tid; i < BB * 16; i += 256) {
        int bi = i >> 4, j = i & 15;
        float ig = gsh[0][bi][j], fg = gsh[1][bi][j];
        float gg = gsh[2][bi][j], og = gsh[3][bi][j];
        int idx = bi * DUNITS + j0 + j;
        float cp = cprev[idx], hp = hprev[idx];
        float c2 = sigmoidf(fg) * cp + sigmoidf(ig) * tanhf(gg);
        float h2 = sigmoidf(og) * tanhf(c2);
        hout[idx] = ZONEOUT * hp + (1.0f - ZONEOUT) * h2;
        cout[idx] = ZONEOUT * cp + (1.0f - ZONEOUT) * c2;
    }
}

// ---------------------------------------------------------------------------
// output projection for all batches at once: zcs=[z1n | att_c] (32 x 1024)
// @ Wf (1024 x 80) via WMMA (2 m-tiles x 5 n-tiles = 10 waves), + stop logit.
// grid: 1 block, 320 threads.
// ---------------------------------------------------------------------------
__global__ void k_out(const float* __restrict__ z1n, const float* __restrict__ x,
                      const float* __restrict__ Wf, const float* __restrict__ bf,
                      const float* __restrict__ Wq, const float* __restrict__ bq,
                      float* __restrict__ out_before, float* __restrict__ out_logits,
                      int t) {
    int tid = threadIdx.x;
    int lane = tid & 31, w = tid >> 5;  // 10 waves
    int l16 = lane & 15, hi = lane >> 4;
    int nt = w >> 1, mt = w & 1;
    int n0 = nt * 16, m0 = mt * 16;
    int m = m0 + l16;  // batch row for A fetch

    v8f acc = {};
    for (int k = 0; k < DUNITS + IDIM; k += 4) {
        int q = k + 2 * hi;
        const float* ap = (q < DUNITS) ? (z1n + m * DUNITS + q)
                                       : (x + m * (IDIM + PUNITS) + (q - DUNITS));
        v2f a; a.x = ap[0]; a.y = ap[1];
        v2f bm; bm.x = Wf[q * ODIM + n0 + l16];
        bm.y = Wf[(q + 1) * ODIM + n0 + l16];
        acc = wmma_f32(a, bm, acc);
    }
    float bias = bf[n0 + l16];
    for (int v = 0; v < 8; v++) {
        int mb = m0 + v + 8 * hi;  // batch index
        out_before[(size_t)(mb * LL + t) * ODIM + n0 + l16] = acc[v] + bias;
    }
    if (tid < BB) {  // wave 0: stop-token logits
        int b = tid;
        float s = bq[0];
        for (int q = 0; q < DUNITS; q++) s += z1n[b * DUNITS + q] * Wq[q];
        for (int q = 0; q < IDIM; q++)
            s += x[b * (IDIM + PUNITS) + q] * Wq[DUNITS + q];
        out_logits[b * LL + t] = s;
    }
}

// ---------------------------------------------------------------------------
// postnet conv layers as implicit-im2col WMMA GEMMs.
// A = conv weights (C_out x C_in*5, contiguous); B = shifted input columns.
// ---------------------------------------------------------------------------

// layer 0: before (B,L,80) -> h0 (B,512,L); K = 400.  grid (64, B), 256 thr.
__global__ void k_pn0(const float* __restrict__ before, const float* __restrict__ pk0,
                      const float* __restrict__ pb0, const float* __restrict__ g0,
                      const float* __restrict__ b0, float* __restrict__ h0) {
    int b = blockIdx.y, tid = threadIdx.x;
    int lane = tid & 31, w = tid >> 5;
    int id = blockIdx.x * 8 + w;        // 512 tiles: 32 co-tiles x 16 l-tiles
    int lt = id & 15, ct = id >> 4;
    int co0 = ct * 16, l0 = lt * 16;
    int l16 = lane & 15, hi = lane >> 4;
    int l = l0 + l16;

    v8f acc = {};
    const float* Arow = pk0 + (size_t)(co0 + l16) * (ODIM * PN_K);
    for (int k = 0; k < ODIM * PN_K; k += 4) {
        int kq = k + 2 * hi;
        v2f a; a.x = Arow[kq]; a.y = Arow[kq + 1];
        int ci0 = kq / 5, kf0 = kq - ci0 * 5;
        int ci1 = (kq + 1) / 5, kf1 = (kq + 1) - ci1 * 5;
        int p0 = l + kf0 - 2, p1 = l + kf1 - 2;
        v2f bm;
        bm.x = (p0 >= 0 && p0 < LL) ? before[(size_t)(b * LL + p0) * ODIM + ci0] : 0.0f;
        bm.y = (p1 >= 0 && p1 < LL) ? before[(size_t)(b * LL + p1) * ODIM + ci1] : 0.0f;
        acc = wmma_f32(a, bm, acc);
    }
    for (int v = 0; v < 8; v++) {
        int co = co0 + v + 8 * hi;
        float val = tanhf(g0[co] * ((acc[v] + pb0[co]) * BN_INV) + b0[co]);
        h0[(size_t)(b * PN_CHANS + co) * LL + l] = val;
    }
}

// layer 1: h0 (B,512,L) -> h1 (B,512,L); K = 2560.  grid (64, B), 256 thr.
__global__ void k_pn1(const float* __restrict__ h0, const float* __restrict__ pk1,
                      const float* __restrict__ pb1, const float* __restrict__ g1,
                      const float* __restrict__ b1, float* __restrict__ h1) {
    int b = blockIdx.y, tid = threadIdx.x;
    int lane = tid & 31, w = tid >> 5;
    int id = blockIdx.x * 8 + w;
    int lt = id & 15, ct = id >> 4;
    int co0 = ct * 16, l0 = lt * 16;
    int l16 = lane & 15, hi = lane >> 4;
    int l = l0 + l16;
    const float* hin = h0 + (size_t)b * PN_CHANS * LL;

    v8f acc = {};
    const float* Arow = pk1 + (size_t)(co0 + l16) * (PN_CHANS * PN_K);
    for (int k = 0; k < PN_CHANS * PN_K; k += 4) {
        int kq = k + 2 * hi;
        v2f a; a.x = Arow[kq]; a.y = Arow[kq + 1];
        int ci0 = kq / 5, kf0 = kq - ci0 * 5;
        int ci1 = (kq + 1) / 5, kf1 = (kq + 1) - ci1 * 5;
        int p0 = l + kf0 - 2, p1 = l + kf1 - 2;
        v2f bm;
        bm.x = (p0 >= 0 && p0 < LL) ? hin[ci0 * LL + p0] : 0.0f;
        bm.y = (p1 >= 0 && p1 < LL) ? hin[ci1 * LL + p1] : 0.0f;
        acc = wmma_f32(a, bm, acc);
    }
    for (int v = 0; v < 8; v++) {
        int co = co0 + v + 8 * hi;
        float val = tanhf(g1[co] * ((acc[v] + pb1[co]) * BN_INV) + b1[co]);
        h1[(size_t)(b * PN_CHANS + co) * LL + l] = val;
    }
}

// layer 2: h1 (B,512,L) -> post; after = before + post; K = 2560.
// grid (10, B), 256 thr: 80 tiles = 5 co-tiles x 16 l-tiles per batch.
__global__ void k_pn2(const float* __restrict__ h1, const float* __restrict__ pk2,
                      const float* __restrict__ pb2, const float* __restrict__ g2,
                      const float* __restrict__ b2,
                      const float* __restrict__ before, float* __restrict__ after) {
    int b = blockIdx.y, tid = threadIdx.x;
    int lane = tid & 31, w = tid >> 5;
    int id = blockIdx.x * 8 + w;        // 0..79
    int lt = id & 15, ct = id >> 4;     // 16 l-tiles, 5 co-tiles
    int co0 = ct * 16, l0 = lt * 16;
    int l16 = lane & 15, hi = lane >> 4;
    int l = l0 + l16;
    const float* hin = h1 + (size_t)b * PN_CHANS * LL;

    v8f acc = {};
    const float* Arow = pk2 + (size_t)(co0 + l16) * (PN_CHANS * PN_K);
    for (int k = 0; k < PN_CHANS * PN_K; k += 4) {
        int kq = k + 2 * hi;
        v2f a; a.x = Arow[kq]; a.y = Arow[kq + 1];
        int ci0 = kq / 5, kf0 = kq - ci0 * 5;
        int ci1 = (kq + 1) / 5, kf1 = (kq + 1) - ci1 * 5;
        int p0 = l + kf0 - 2, p1 = l + kf1 - 2;
        v2f bm;
        bm.x = (p0 >= 0 && p0 < LL) ? hin[ci0 * LL + p0] : 0.0f;
        bm.y = (p1 >= 0 && p1 < LL) ? hin[ci1 * LL + p1] : 0.0f;
        acc = wmma_f32(a, bm, acc);
    }
    for (int v = 0; v < 8; v++) {
        int co = co0 + v + 8 * hi;
        float post = g2[co] * ((acc[v] + pb2[co]) * BN_INV) + b2[co];
        size_t idx = (size_t)(b * LL + l) * ODIM + co;
        after[idx] = before[idx] + post;
    }
}

// ---------------------------------------------------------------------------
extern "C" void kernel_launch(void* const* d_in, const int* in_sizes, int n_in,
                              void* d_out, int out_size, void* d_ws, size_t ws_size,
                              hipStream_t stream) {
    const float* enc_z = (const float*)d_in[0];
    const float* ys    = (const float*)d_in[1];
    // d_in[2] = spk_embeds (unused by the reference)
    const int*   ilens = (const int*)d_in[3];
    const float* W_enc = (const float*)d_in[4];
    const float* b_enc = (const float*)d_in[5];
    const float* W_dec = (const float*)d_in[6];
    const float* loc_k = (const float*)d_in[7];
    const float* W_att = (const float*)d_in[8];
    const float* g_w   = (const float*)d_in[9];
    const float* g_b   = (const float*)d_in[10];
    const float* Wp0   = (const float*)d_in[11];
    const float* bp0   = (const float*)d_in[12];
    const float* Wp1   = (const float*)d_in[13];
    const float* bp1   = (const float*)d_in[14];
    const float* Wih0  = (const float*)d_in[15];
    const float* Whh0  = (const float*)d_in[16];
    const float* bih0  = (const float*)d_in[17];
    const float* bhh0  = (const float*)d_in[18];
    const float* Wih1  = (const float*)d_in[19];
    const float* Whh1  = (const float*)d_in[20];
    const float* bih1  = (const float*)d_in[21];
    const float* bhh1  = (const float*)d_in[22];
    const float* Wf    = (const float*)d_in[23];
    const float* bf    = (const float*)d_in[24];
    const float* Wq    = (const float*)d_in[25];
    const float* bq    = (const float*)d_in[26];
    const float* pk0   = (const float*)d_in[27];
    const float* pb0   = (const float*)d_in[28];
    const float* bn_g0 = (const float*)d_in[29];
    const float* bn_b0 = (const float*)d_in[30];
    const float* pk1   = (const float*)d_in[31];
    const float* pb1   = (const float*)d_in[32];
    const float* bn_g1 = (const float*)d_in[33];
    const float* bn_b1 = (const float*)d_in[34];
    const float* pk2   = (const float*)d_in[35];
    const float* pb2   = (const float*)d_in[36];
    const float* bn_g2 = (const float*)d_in[37];
    const float* bn_b2 = (const float*)d_in[38];

    float* ws  = (float*)d_ws;
    float* out = (float*)d_out;
    float* out_after  = out + OUT_AFTER;
    float* out_before = out + OUT_BEFORE;
    float* out_logits = out + OUT_LOGITS;
    float* out_aws    = out + OUT_AWS;

    k_init<<<64, 256, 0, stream>>>(ws, ilens);
    k_enc_proj<<<512, 256, 0, stream>>>(enc_z, W_enc, b_enc, ws + WS_ENCPROJ);

    const int SLOT = BB * DUNITS;  // 16384
    for (int t = 0; t < LL; t++) {
        int s = t & 1;
        k_att<<<BB, 256, 0, stream>>>(enc_z, ys, ilens, W_dec, loc_k, W_att, g_w, g_b,
                                      Wp0, bp0, Wp1, bp1, ws, out_aws, t);
        k_lstm<<<32, 256, 0, stream>>>(
            ws + WS_X, IDIM + PUNITS, Wih0, ws + WS_Z0 + s * SLOT, Whh0, bih0, bhh0,
            ws + WS_C0 + s * SLOT, ws + WS_Z0 + (s ^ 1) * SLOT,
            ws + WS_C0 + (s ^ 1) * SLOT);
        k_lstm<<<32, 256, 0, stream>>>(
            ws + WS_Z0 + (s ^ 1) * SLOT, DUNITS, Wih1, ws + WS_Z1 + s * SLOT, Whh1,
            bih1, bhh1, ws + WS_C1 + s * SLOT, ws + WS_Z1 + (s ^ 1) * SLOT,
            ws + WS_C1 + (s ^ 1) * SLOT);
        k_out<<<1, 320, 0, stream>>>(ws + WS_Z1 + (s ^ 1) * SLOT, ws + WS_X, Wf, bf,
                                     Wq, bq, out_before, out_logits, t);
    }

    k_pn0<<<dim3(64, BB), 256, 0, stream>>>(out_before, pk0, pb0, bn_g0, bn_b0,
                                            ws + WS_H0);
    k_pn1<<<dim3(64, BB), 256, 0, stream>>>(ws + WS_H0, pk1, pb1, bn_g1, bn_b1,
                                            ws + WS_H1);
    k_pn2<<<dim3(10, BB), 256, 0, stream>>>(ws + WS_H1, pk2, pb2, bn_g2, bn_b2,
                                            out_before, out_after);
}